// GNNStack_687194767739
// MI455X (gfx1250) — compile-verified
//
#include <hip/hip_runtime.h>
#include <hip/hip_bf16.h>
#include <math.h>

#define CHAN 128
#define LDSW 132   // padded row stride (132 mod 64 = 4 -> conflict-free half-wave reads)

typedef __attribute__((ext_vector_type(2))) float v2f;
typedef __attribute__((ext_vector_type(8))) float v8f;

// ---------------------------------------------------------------------------
// Kernel 1: per-node degree -> 1/sqrt(deg).  One wave32 per node (K<=32).
// ---------------------------------------------------------------------------
__global__ __launch_bounds__(256) void deg_kernel(const long long* __restrict__ ei,
                                                  float* __restrict__ inv_sqrt,
                                                  int n_nodes, int K) {
    int gid  = blockIdx.x * blockDim.x + threadIdx.x;
    int node = gid >> 5;
    int lane = threadIdx.x & 31;
    if (node >= n_nodes) return;
    long long idx = (lane < K) ? ei[(size_t)node * K + lane] : (long long)-1;
    unsigned long long m = __ballot(idx >= 0);
    int deg = __popcll(m) + 1;   // valid neighbors + self
    if (lane == 0) inv_sqrt[node] = rsqrtf((float)deg);
}

// ---------------------------------------------------------------------------
// Kernel 2: h[r, :] = (x[r, :] * inv_sqrt[r]) @ W^T      (scale fused into A)
// Block = 256 threads = 8 waves sharing one 16-row x tile; W (128x128) and
// the pre-scaled x tile staged in LDS with 132-float padded rows.  Wave w
// computes the 16x16 tile at columns [16w,16w+16) via 32 chained
// v_wmma_f32_16x16x4_f32 ops over K=128.
//
// f32 16x16x4 fragment layout (wave32):
//   A (16x4):  lane l<16: A[M=l][K=0..1] in v0..v1; lane l+16: K=2..3.
//   B (4x16):  lane l<16: B[K=0..1][N=l];           lane l+16: K=2..3.
//   C/D:       VGPR i: lanes 0-15 -> M=i, lanes 16-31 -> M=i+8; N=lane%16.
// ---------------------------------------------------------------------------
__global__ __launch_bounds__(256) void gemm_scale_kernel(const float* __restrict__ x,
                                                         const float* __restrict__ W,
                                                         const float* __restrict__ inv_sqrt,
                                                         float* __restrict__ h,
                                                         int n_nodes) {
    __shared__ float sW[CHAN * LDSW];   // ~66 KB: W[c_out][c_in], padded rows
    __shared__ float sX[16 * LDSW];     // ~8.3 KB: pre-scaled 16-row x tile

    const int t    = threadIdx.x;
    const int row0 = blockIdx.x * 16;
    const bool full = (row0 + 16 <= n_nodes);

    // Stage W: 4096 float4s -> padded rows (row = i/32, col4 = i%32).
    const float4* W4 = (const float4*)W;
    #pragma unroll 4
    for (int i = t; i < (CHAN * CHAN) / 4; i += 256) {
        *(float4*)(sW + (i >> 5) * LDSW + (i & 31) * 4) = W4[i];
    }

    // Stage x tile (512 float4s), pre-scaled by inv_sqrt of its row.
    const float4* X4 = (const float4*)(x + (size_t)row0 * CHAN);
    if (full) {
        #pragma unroll 2
        for (int i = t; i < (16 * CHAN) / 4; i += 256) {
            float  s = inv_sqrt[row0 + (i >> 5)];
            float4 v = X4[i];
            v.x *= s; v.y *= s; v.z *= s; v.w *= s;
            *(float4*)(sX + (i >> 5) * LDSW + (i & 31) * 4) = v;
        }
    } else {
        for (int i = t; i < (16 * CHAN) / 4; i += 256) {
            int r = row0 + (i >> 5);
            float4 v = make_float4(0.f, 0.f, 0.f, 0.f);
            if (r < n_nodes) {
                float s = inv_sqrt[r];
                v = X4[i];
                v.x *= s; v.y *= s; v.z *= s; v.w *= s;
            }
            *(float4*)(sX + (i >> 5) * LDSW + (i & 31) * 4) = v;
        }
    }
    __syncthreads();

    const int wave = t >> 5;          // column tile 0..7
    const int lane = t & 31;
    const int half = lane >> 4;       // K pair {0,1} vs {2,3}
    const int lm   = lane & 15;
    const int col0 = wave * 16;

    const float* pa = sX + lm * LDSW + 2 * half;           // A row lm
    const float* pb = sW + (col0 + lm) * LDSW + 2 * half;  // B[k][n] = W[col0+lm][k]

    v8f acc = {};
    #pragma unroll 8
    for (int kk = 0; kk < CHAN / 4; ++kk) {           // k0 = 4*kk
        v2f a = *(const v2f*)(pa + 4 * kk);
        v2f b = *(const v2f*)(pb + 4 * kk);
        acc = __builtin_amdgcn_wmma_f32_16x16x4_f32(
            false, a, false, b, (short)0, acc, false, false);
    }

    // Straight-line stores (scale already folded into A).
    float* hp = h + (size_t)(row0 + 8 * half) * CHAN + col0 + lm;
    if (full) {
        #pragma unroll
        for (int i = 0; i < 8; ++i)
            hp[(size_t)i * CHAN] = acc[i];
    } else {
        #pragma unroll
        for (int i = 0; i < 8; ++i) {
            int r = row0 + i + 8 * half;
            if (r < n_nodes) hp[(size_t)i * CHAN] = acc[i];
        }
    }
}

// ---------------------------------------------------------------------------
// Kernel 3: out[n,:] = elu( (sum_j h[ei[n,j],:] + h[n,:]) * inv_sqrt[n] )
// One wave per node; lane owns 4 channels (float4) -> coalesced 512B row
// reads.  Neighbor indices broadcast with __shfl so the per-j branch is
// wave-uniform.  This phase is L2-bandwidth bound (~1.6 GB of gathered rows
// per layer; h fits in the 192MB L2).
// ---------------------------------------------------------------------------
__global__ __launch_bounds__(256) void agg_elu_kernel(const float* __restrict__ h,
                                                      const long long* __restrict__ ei,
                                                      const float* __restrict__ inv_sqrt,
                                                      float* __restrict__ out,
                                                      int n_nodes, int K) {
    int gid  = blockIdx.x * blockDim.x + threadIdx.x;
    int node = gid >> 5;
    int lane = threadIdx.x & 31;
    if (node >= n_nodes) return;

    long long myidx = (lane < K) ? ei[(size_t)node * K + lane] : (long long)-1;

    // self term
    const float4* hv = (const float4*)(h + (size_t)node * CHAN);
    float4 acc = hv[lane];

    #pragma unroll 4
    for (int j = 0; j < 32; ++j) {
        if (j >= K) break;
        long long idx = __shfl(myidx, j, 32);
        if (idx >= 0) {
            const float4 v = *(const float4*)(h + (size_t)idx * CHAN + lane * 4);
            acc.x += v.x; acc.y += v.y; acc.z += v.z; acc.w += v.w;
        }
    }

    float s = inv_sqrt[node];
    float r0 = acc.x * s, r1 = acc.y * s, r2 = acc.z * s, r3 = acc.w * s;
    r0 = r0 > 0.f ? r0 : expm1f(r0);
    r1 = r1 > 0.f ? r1 : expm1f(r1);
    r2 = r2 > 0.f ? r2 : expm1f(r2);
    r3 = r3 > 0.f ? r3 : expm1f(r3);
    float4 res = make_float4(r0, r1, r2, r3);
    *(float4*)(out + (size_t)node * CHAN + lane * 4) = res;
}

// ---------------------------------------------------------------------------
// Launcher: deg -> [gemm -> agg] x 2
// inputs: x[N*128] f32, edge_index[N*K] i64, W1[128*128] f32, W2[128*128] f32
// ws layout (floats): inv_sqrt[N] | h[N*128] | y1[N*128]
// ---------------------------------------------------------------------------
extern "C" void kernel_launch(void* const* d_in, const int* in_sizes, int n_in,
                              void* d_out, int out_size, void* d_ws, size_t ws_size,
                              hipStream_t stream) {
    const float*     x  = (const float*)d_in[0];
    const long long* ei = (const long long*)d_in[1];
    const float*     W1 = (const float*)d_in[2];
    const float*     W2 = (const float*)d_in[3];
    float* out = (float*)d_out;

    const int N = in_sizes[0] / CHAN;
    const int K = in_sizes[1] / N;

    float* inv_sqrt = (float*)d_ws;
    size_t off_h = (((size_t)N + 63) & ~(size_t)63);   // 256B-align
    float* h  = inv_sqrt + off_h;
    float* y1 = h + (size_t)N * CHAN;

    const int waveBlocks = (int)(((size_t)N * 32 + 255) / 256);
    const int mtiles     = (N + 15) / 16;

    deg_kernel<<<waveBlocks, 256, 0, stream>>>(ei, inv_sqrt, N, K);

    // Layer 1
    gemm_scale_kernel<<<mtiles, 256, 0, stream>>>(x, W1, inv_sqrt, h, N);
    agg_elu_kernel<<<waveBlocks, 256, 0, stream>>>(h, ei, inv_sqrt, y1, N, K);

    // Layer 2
    gemm_scale_kernel<<<mtiles, 256, 0, stream>>>(y1, W2, inv_sqrt, h, N);
    agg_elu_kernel<<<waveBlocks, 256, 0, stream>>>(h, ei, inv_sqrt, out, N, K);
}